// ThreeBodyEquiGraphConvSimple_31499290149200
// MI455X (gfx1250) — compile-verified
//
#include <hip/hip_runtime.h>
#include <hip/hip_bf16.h>

// ---------------------------------------------------------------------------
// Problem constants (from reference)
// ---------------------------------------------------------------------------
#define N_NODES  50000
#define N_EDGES  1600000
#define G_DIM    128
#define LG       16
#define N_TILES_NODE (N_NODES / 16)   // 3125
#define N_TILES_EDGE (N_EDGES / 16)   // 100000
#define PI_OVER_CUTOFF 0.314159265358979f   // pi / 10
// workspace layout (floats)
#define WS_NS_OFF   0L
#define WS_NV_OFF   800000L     // N*LG
#define WS_NES_OFF  3200000L    // + N*3*LG
#define WS_NEV_OFF  4000000L
// output layout (floats)
#define OUT_NS_OFF  0L
#define OUT_NV_OFF  6400000L    // N*G
#define OUT_ES_OFF  25600000L   // + N*3*G
#define OUT_EV_OFF  51200000L   // + E*LG

typedef __attribute__((ext_vector_type(2))) float v2f;
typedef __attribute__((ext_vector_type(8))) float v8f;

__device__ __forceinline__ void lds_fence() {
    // cross-lane LDS RAW/WAR round-trips within a wave: DS ops are in-order per
    // wave, this forces the wait + stops compiler reordering.
    asm volatile("s_wait_dscnt 0x0" ::: "memory");
}

__device__ __forceinline__ float siluf(float x) { return x * (1.0f / (1.0f + __expf(-x))); }
__device__ __forceinline__ float sigmf(float x) { return 1.0f / (1.0f + __expf(-x)); }

__device__ __forceinline__ void atomAddF(float* p, float v) {
    unsafeAtomicAdd(p, v);   // global_atomic_add_f32 (no return)
}

// ---------------------------------------------------------------------------
// f32 WMMA helpers (V_WMMA_F32_16X16X4_F32)
// D = A(16xK, LDS tile) x W(KxN row-major; cols [colOff,colOff+16)) + C
// A operand layout (ISA 32-bit A 16x4): lane m=lane&15, hi=lane>>4,
//   a[d] = A[m][k + 2*hi + d]
// B operand (ISA B row-striping): b[d] = W[k + 2*hi + d][n], n = lane&15
// C/D layout: d[r] = D[r + 8*hi][lane&15]
// ---------------------------------------------------------------------------
template <int K>
__device__ __forceinline__ void load_Bops(const float* W, int ldw, int colOff,
                                          v2f* B, int lane) {
    const int n  = lane & 15;
    const int hi = lane >> 4;
#pragma unroll
    for (int k = 0; k < K; k += 4) {
        B[k / 4][0] = W[(k + 2 * hi + 0) * ldw + colOff + n];
        B[k / 4][1] = W[(k + 2 * hi + 1) * ldw + colOff + n];
    }
}

template <int K>
__device__ __forceinline__ v8f wmma_pre(const float* A, int lda, const v2f* B,
                                        v8f c, int lane) {
    const int m  = lane & 15;
    const int hi = lane >> 4;
#pragma unroll
    for (int k = 0; k < K; k += 4) {
        v2f a;
        a[0] = A[m * lda + k + 2 * hi + 0];
        a[1] = A[m * lda + k + 2 * hi + 1];
        c = __builtin_amdgcn_wmma_f32_16x16x4_f32(false, a, false, B[k / 4],
                                                  (short)0, c, false, false);
    }
    return c;
}

__device__ __forceinline__ v8f splat8(float v) {
    v8f c;
#pragma unroll
    for (int r = 0; r < 8; ++r) c[r] = v;
    return c;
}

// ---------------------------------------------------------------------------
// Kernel 0: zero the scatter accumulators
// ---------------------------------------------------------------------------
__global__ void zero_kernel(float* __restrict__ p, long n) {
    const long idx = (long)blockIdx.x * blockDim.x + threadIdx.x;
    if (idx < n) p[idx] = 0.0f;
}

// ---------------------------------------------------------------------------
// Kernel 1a: n_s = silu(node_s @ ns_in_w + b)   [N,16]
// Persistent waves, weights preloaded in VGPRs (32 K-steps = 64 VGPRs).
// ---------------------------------------------------------------------------
__global__ __launch_bounds__(128) void node_in_s_kernel(
    const float* __restrict__ node_s,
    const float* __restrict__ ns_w, const float* __restrict__ ns_b,
    float* __restrict__ n_s) {
    __shared__ float tile[4][16][132];
    const int wave = threadIdx.x >> 5;
    const int lane = threadIdx.x & 31;
    float(*A)[132] = tile[wave];
    const int col = lane & 15;
    const int hi  = lane >> 4;

    v2f wB[32];
    load_Bops<128>(ns_w, 16, 0, wB, lane);
    const float bias = ns_b[col];

    const int stride = gridDim.x * 4;
    for (int t = blockIdx.x * 4 + wave; t < N_TILES_NODE; t += stride) {
        const int n0 = t * 16;
        lds_fence();   // WAR vs previous iteration A reads
#pragma unroll 4
        for (int r = 0; r < 16; ++r) {
            const float4 v = ((const float4*)(node_s + (long)(n0 + r) * G_DIM))[lane];
            *((float4*)&A[r][lane * 4]) = v;
        }
        lds_fence();
        v8f c = splat8(bias);
        c = wmma_pre<128>(&A[0][0], 132, wB, c, lane);
#pragma unroll
        for (int r = 0; r < 8; ++r)
            n_s[(long)(n0 + r + 8 * hi) * LG + col] = siluf(c[r]);
    }
}

// ---------------------------------------------------------------------------
// Kernel 1b: n_v = node_v @ nv_in_w   [N,3,16]
// ---------------------------------------------------------------------------
__global__ __launch_bounds__(128) void node_in_v_kernel(
    const float* __restrict__ node_v, const float* __restrict__ nv_w,
    float* __restrict__ n_v) {
    __shared__ float tile[4][16][132];
    const int wave = threadIdx.x >> 5;
    const int lane = threadIdx.x & 31;
    float(*A)[132] = tile[wave];
    const int col = lane & 15;
    const int hi  = lane >> 4;

    v2f wB[32];
    load_Bops<128>(nv_w, 16, 0, wB, lane);

    const int stride = gridDim.x * 4;
    for (int t = blockIdx.x * 4 + wave; t < N_TILES_NODE; t += stride) {
        const int n0 = t * 16;
        for (int x = 0; x < 3; ++x) {
            lds_fence();
#pragma unroll 4
            for (int r = 0; r < 16; ++r) {
                const float4 v =
                    ((const float4*)(node_v + ((long)(n0 + r) * 3 + x) * G_DIM))[lane];
                *((float4*)&A[r][lane * 4]) = v;
            }
            lds_fence();
            v8f cv = {};
            cv = wmma_pre<128>(&A[0][0], 132, wB, cv, lane);
#pragma unroll
            for (int r = 0; r < 8; ++r)
                n_v[((long)(n0 + r + 8 * hi) * 3 + x) * LG + col] = cv[r];
        }
    }
}

// ---------------------------------------------------------------------------
// Kernel 2: per-edge update. One wave = 16-edge tile; persistent over tiles,
// all weight B fragments live in VGPRs (32 x v2f = 64 VGPRs).
// ---------------------------------------------------------------------------
__global__ __launch_bounds__(256) void edge_kernel(
    const float* __restrict__ edge_s_in, const float* __restrict__ edge_v_in,
    const float* __restrict__ dist, const float* __restrict__ vctr,
    const int* __restrict__ src, const int* __restrict__ dst,
    const float* __restrict__ n_s, const float* __restrict__ n_v,
    const float* __restrict__ en_w, const float* __restrict__ en_b,
    const float* __restrict__ tp_w, const float* __restrict__ tp_b,
    const float* __restrict__ gw1, const float* __restrict__ gb1,
    const float* __restrict__ gw2, const float* __restrict__ gb2,
    const float* __restrict__ tv_w, const float* __restrict__ tv_b,
    float* __restrict__ n_es, float* __restrict__ n_ev,
    float* __restrict__ edge_s_out, float* __restrict__ edge_v_out) {
    struct Tile {
        float cat[16][33];   // [n_s[src] | n_s[dst]]  (16 x 32)
        float es[16][17];    // edge_s tile
        float t1[16][17];    // D-layout -> A-layout staging
        float Cc[16];        // cosine cutoff
        float vn[16][4];     // vctr_norm
        int   sidx[16];
        int   didx[16];
    };
    __shared__ Tile tiles[8];
    const int wave = threadIdx.x >> 5;
    const int lane = threadIdx.x & 31;
    Tile& T        = tiles[wave];
    const int i    = lane & 15;    // edge-in-tile (also output column)
    const int hi   = lane >> 4;

    // ---- loop-invariant weights / biases in registers ----
    v2f wEn[8], wTp[4], wG1[4], wG2[4], wTt[4], wTe[4], wTr[4];
    load_Bops<32>(en_w, 16, 0, wEn, lane);
    load_Bops<16>(tp_w, 16, 0, wTp, lane);
    load_Bops<16>(gw1, 16, 0, wG1, lane);
    load_Bops<16>(gw2, 16, 0, wG2, lane);
    load_Bops<16>(tv_w, 48, 0, wTt, lane);
    load_Bops<16>(tv_w, 48, 16, wTe, lane);
    load_Bops<16>(tv_w, 48, 32, wTr, lane);
    const float bEn = en_b[i], bTp = tp_b[i], bG1 = gb1[i], bG2 = gb2[i];
    const float bTt = tv_b[i], bTe = tv_b[16 + i], bTr = tv_b[32 + i];

    const long stride = (long)gridDim.x * 8;
    for (long tile = (long)blockIdx.x * 8 + wave; tile < N_TILES_EDGE;
         tile += stride) {
        const long e0  = tile * 16;
        const long e   = e0 + i;
        const int  s_i = src[e];
        const int  d_i = dst[e];

        lds_fence();   // WAR vs previous iteration LDS reads
        if (hi == 0) {
            T.sidx[i] = s_i;
            T.didx[i] = d_i;
            const float d = dist[e];
            T.Cc[i] = (d < 10.0f) ? 0.5f * (__cosf(PI_OVER_CUTOFF * d) + 1.0f) : 0.0f;
            T.vn[i][0] = vctr[e * 3 + 0];
            T.vn[i][1] = vctr[e * 3 + 1];
            T.vn[i][2] = vctr[e * 3 + 2];
        }
        {   // gather node scalars: half-wave 0 -> src rows, half-wave 1 -> dst
            const int     nn  = hi ? d_i : s_i;
            const float4* row = (const float4*)(n_s + (long)nn * LG);
            const float4 a0 = row[0], a1 = row[1], a2 = row[2], a3 = row[3];
            float* p = &T.cat[i][hi * 16];
            p[0] = a0.x; p[1] = a0.y; p[2]  = a0.z; p[3]  = a0.w;
            p[4] = a1.x; p[5] = a1.y; p[6]  = a1.z; p[7]  = a1.w;
            p[8] = a2.x; p[9] = a2.y; p[10] = a2.z; p[11] = a2.w;
            p[12] = a3.x; p[13] = a3.y; p[14] = a3.z; p[15] = a3.w;
        }
        {   // edge_s tile
            const float4* row = (const float4*)(edge_s_in + e * LG + hi * 8);
            const float4 b0 = row[0], b1 = row[1];
            float* p = &T.es[i][hi * 8];
            p[0] = b0.x; p[1] = b0.y; p[2] = b0.z; p[3] = b0.w;
            p[4] = b1.x; p[5] = b1.y; p[6] = b1.z; p[7] = b1.w;
        }
        lds_fence();

        // edge_neighbors = node_cat @ en_w + b ; tp = edge_s @ tp_w + b
        v8f Den = splat8(bEn);
        Den = wmma_pre<32>(&T.cat[0][0], 33, wEn, Den, lane);
        v8f Dtp = splat8(bTp);
        Dtp = wmma_pre<16>(&T.es[0][0], 17, wTp, Dtp, lane);
        v8f Tm;
#pragma unroll
        for (int r = 0; r < 8; ++r) Tm[r] = Den[r] * Dtp[r];   // triplet_message

        // gate = sigmoid(silu(Tm@gw1+b1)@gw2+b2)
#pragma unroll
        for (int r = 0; r < 8; ++r) T.t1[r + 8 * hi][i] = Tm[r];
        lds_fence();
        v8f H = splat8(bG1);
        H = wmma_pre<16>(&T.t1[0][0], 17, wG1, H, lane);
#pragma unroll
        for (int r = 0; r < 8; ++r) H[r] = siluf(H[r]);
        lds_fence();   // WAR on t1
#pragma unroll
        for (int r = 0; r < 8; ++r) T.t1[r + 8 * hi][i] = H[r];
        lds_fence();
        v8f G = splat8(bG2);
        G = wmma_pre<16>(&T.t1[0][0], 17, wG2, G, lane);

        float Cr[8];
#pragma unroll
        for (int r = 0; r < 8; ++r) Cr[r] = T.Cc[r + 8 * hi];
        v8f S;   // e_s_upd
#pragma unroll
        for (int r = 0; r < 8; ++r) S[r] = Tm[r] * sigmf(G[r]) * Cr[r];

        // vec_ch = e_s_upd @ tv_w + tv_b  -> t/e/r channel blocks
        lds_fence();
#pragma unroll
        for (int r = 0; r < 8; ++r) T.t1[r + 8 * hi][i] = S[r];
        lds_fence();
        v8f Dt = splat8(bTt);
        Dt = wmma_pre<16>(&T.t1[0][0], 17, wTt, Dt, lane);
        v8f De = splat8(bTe);
        De = wmma_pre<16>(&T.t1[0][0], 17, wTe, De, lane);
        v8f Dr = splat8(bTr);
        Dr = wmma_pre<16>(&T.t1[0][0], 17, wTr, Dr, lane);

        // scalar outputs + scatter
#pragma unroll
        for (int r = 0; r < 8; ++r) {
            const int   row = r + 8 * hi;
            const long  ee  = e0 + row;
            const float sv  = S[r];
            edge_s_out[ee * LG + i] = sv + T.es[row][i];
            atomAddF(&n_es[(long)T.didx[row] * LG + i], sv);
        }
        // vector outputs + scatter
#pragma unroll
        for (int x = 0; x < 3; ++x) {
#pragma unroll
            for (int r = 0; r < 8; ++r) {
                const int   row = r + 8 * hi;
                const long  ee  = e0 + row;
                const long  off = (ee * 3 + x) * LG + i;
                const float ev  = edge_v_in[off];
                const float nv  = n_v[((long)T.sidx[row] * 3 + x) * LG + i];
                const float vnx = T.vn[row][x];
                const float upd = (ev * Dt[r] + nv * De[r] + vnx * Dr[r]) * Cr[r];
                edge_v_out[off] = upd + ev;
                atomAddF(&n_ev[((long)T.didx[row] * 3 + x) * LG + i], upd);
            }
        }
    }
}

// ---------------------------------------------------------------------------
// Kernel 3: node updates (16 nodes per block-iteration, 4 waves; persistent).
// Each wave only ever needs output column blocks j = wave and wave+4.
// ---------------------------------------------------------------------------
__global__ __launch_bounds__(128) void node_out_kernel(
    const float* __restrict__ node_s, const float* __restrict__ node_v,
    const float* __restrict__ n_v, const float* __restrict__ n_es,
    const float* __restrict__ n_ev, const float* __restrict__ onv_w,
    const float* __restrict__ ow1, const float* __restrict__ ob1,
    const float* __restrict__ ow2, const float* __restrict__ ob2,
    const float* __restrict__ gamma, const float* __restrict__ beta,
    const float* __restrict__ cscale,
    float* __restrict__ node_s_out, float* __restrict__ node_v_out) {
    __shared__ float mess[16][33];        // [n_es | n_v_norm]
    __shared__ float htile[16][17];
    __shared__ float vtile[3][16][20];    // n_ev tile
    __shared__ float supd[16][132];       // n_s_upd
    __shared__ float vupd[3][16][132];    // n_v_upd
    __shared__ float redS[16][8], redQ[16][8], muA[16], rsA[16];

    const int tid  = threadIdx.x;
    const int wave = tid >> 5;
    const int lane = tid & 31;
    const int col  = lane & 15;
    const int hi   = lane >> 4;

    // loop-invariant weight fragments
    const int jA = wave, jB = wave + 4;
    v2f wO1[8], wO2A[4], wO2B[4], wNvA[4], wNvB[4];
    load_Bops<32>(ow1, 16, 0, wO1, lane);
    load_Bops<16>(ow2, 128, 16 * jA, wO2A, lane);
    load_Bops<16>(ow2, 128, 16 * jB, wO2B, lane);
    load_Bops<16>(onv_w, 128, 16 * jA, wNvA, lane);
    load_Bops<16>(onv_w, 128, 16 * jB, wNvB, lane);
    const float b1  = ob1[col];
    const float b2A = ob2[16 * jA + col];
    const float b2B = ob2[16 * jB + col];

    for (int t = blockIdx.x; t < N_TILES_NODE; t += gridDim.x) {
        const int n0 = t * 16;

        for (int w = tid; w < 256; w += 128) {
            const int ii = w >> 4, l = w & 15;
            mess[ii][l] = n_es[(long)(n0 + ii) * LG + l];
            float q = 0.0f;
#pragma unroll
            for (int x = 0; x < 3; ++x) {
                const float v = n_v[((long)(n0 + ii) * 3 + x) * LG + l];
                q += v * v;
            }
            mess[ii][16 + l] = sqrtf(q);
        }
        for (int w = tid; w < 768; w += 128) {
            const int x = w >> 8, rem = w & 255, ii = rem >> 4, l = rem & 15;
            vtile[x][ii][l] = n_ev[((long)(n0 + ii) * 3 + x) * LG + l];
        }
        __syncthreads();

        if (wave == 0) {   // h = silu(n_mess @ ons_w1 + b1)
            v8f h = splat8(b1);
            h = wmma_pre<32>(&mess[0][0], 33, wO1, h, lane);
#pragma unroll
            for (int r = 0; r < 8; ++r) htile[r + 8 * hi][col] = siluf(h[r]);
        }
        __syncthreads();

        {   // n_s_upd = h @ ons_w2 + b2   (this wave's two column blocks)
            v8f c = splat8(b2A);
            c = wmma_pre<16>(&htile[0][0], 17, wO2A, c, lane);
#pragma unroll
            for (int r = 0; r < 8; ++r) supd[r + 8 * hi][16 * jA + col] = c[r];
            c = splat8(b2B);
            c = wmma_pre<16>(&htile[0][0], 17, wO2B, c, lane);
#pragma unroll
            for (int r = 0; r < 8; ++r) supd[r + 8 * hi][16 * jB + col] = c[r];
        }
#pragma unroll
        for (int x = 0; x < 3; ++x) {   // n_v_upd = n_ev @ out_nv_w
            v8f c = {};
            c = wmma_pre<16>(&vtile[x][0][0], 20, wNvA, c, lane);
#pragma unroll
            for (int r = 0; r < 8; ++r) vupd[x][r + 8 * hi][16 * jA + col] = c[r];
            v8f c2 = {};
            c2 = wmma_pre<16>(&vtile[x][0][0], 20, wNvB, c2, lane);
#pragma unroll
            for (int r = 0; r < 8; ++r) vupd[x][r + 8 * hi][16 * jB + col] = c2[r];
        }
        __syncthreads();

        // ---- residual + layernorm ----
        const int ii  = tid >> 3;
        const int sub = tid & 7;
        float xv[16];
        float s = 0.0f, q = 0.0f;
#pragma unroll
        for (int j = 0; j < 16; ++j) {
            const int cch = sub * 16 + j;
            const float x = supd[ii][cch] + node_s[(long)(n0 + ii) * G_DIM + cch];
            xv[j] = x; s += x; q += x * x;
        }
        redS[ii][sub] = s;
        redQ[ii][sub] = q;
        __syncthreads();
        if (sub == 0) {
            float ss = 0.0f, qq = 0.0f;
#pragma unroll
            for (int k = 0; k < 8; ++k) { ss += redS[ii][k]; qq += redQ[ii][k]; }
            const float mu  = ss * (1.0f / 128.0f);
            const float var = qq * (1.0f / 128.0f) - mu * mu;
            muA[ii] = mu;
            rsA[ii] = rsqrtf(var + 1e-5f);
        }
        __syncthreads();
        {
            const float mu = muA[ii], rs = rsA[ii];
#pragma unroll
            for (int j = 0; j < 16; ++j) {
                const int cch = sub * 16 + j;
                node_s_out[(long)(n0 + ii) * G_DIM + cch] =
                    (xv[j] - mu) * rs * gamma[cch] + beta[cch];
            }
        }
        // ---- vector residual + cosine-norm scaling ----
#pragma unroll 2
        for (int j = 0; j < 16; ++j) {
            const int cch = sub * 16 + j;
            const float v0 =
                vupd[0][ii][cch] + node_v[((long)(n0 + ii) * 3 + 0) * G_DIM + cch];
            const float v1 =
                vupd[1][ii][cch] + node_v[((long)(n0 + ii) * 3 + 1) * G_DIM + cch];
            const float v2 =
                vupd[2][ii][cch] + node_v[((long)(n0 + ii) * 3 + 2) * G_DIM + cch];
            const float nrm = sqrtf(v0 * v0 + v1 * v1 + v2 * v2);
            const float sc  = cscale[cch] / (nrm + 1e-8f);
            node_v_out[((long)(n0 + ii) * 3 + 0) * G_DIM + cch] = v0 * sc;
            node_v_out[((long)(n0 + ii) * 3 + 1) * G_DIM + cch] = v1 * sc;
            node_v_out[((long)(n0 + ii) * 3 + 2) * G_DIM + cch] = v2 * sc;
        }
    }
}

// ---------------------------------------------------------------------------
extern "C" void kernel_launch(void* const* d_in, const int* in_sizes, int n_in,
                              void* d_out, int out_size, void* d_ws, size_t ws_size,
                              hipStream_t stream) {
    (void)in_sizes; (void)n_in; (void)out_size; (void)ws_size;
    const float* node_s = (const float*)d_in[0];
    const float* node_v = (const float*)d_in[1];
    const float* edge_s = (const float*)d_in[2];
    const float* edge_v = (const float*)d_in[3];
    const float* dist   = (const float*)d_in[4];
    const float* vctr   = (const float*)d_in[5];
    const int*   src    = (const int*)d_in[6];
    const int*   dst    = (const int*)d_in[7];
    const float* ns_w   = (const float*)d_in[8];
    const float* ns_b   = (const float*)d_in[9];
    const float* nv_w   = (const float*)d_in[10];
    const float* en_w   = (const float*)d_in[11];
    const float* en_b   = (const float*)d_in[12];
    const float* tp_w   = (const float*)d_in[13];
    const float* tp_b   = (const float*)d_in[14];
    const float* gw1    = (const float*)d_in[15];
    const float* gb1    = (const float*)d_in[16];
    const float* gw2    = (const float*)d_in[17];
    const float* gb2    = (const float*)d_in[18];
    const float* tv_w   = (const float*)d_in[19];
    const float* tv_b   = (const float*)d_in[20];
    const float* onv_w  = (const float*)d_in[21];
    const float* ow1    = (const float*)d_in[22];
    const float* ob1    = (const float*)d_in[23];
    const float* ow2    = (const float*)d_in[24];
    const float* ob2    = (const float*)d_in[25];
    const float* gamma  = (const float*)d_in[26];
    const float* beta   = (const float*)d_in[27];
    const float* cscale = (const float*)d_in[28];

    float* ws    = (float*)d_ws;
    float* w_ns  = ws + WS_NS_OFF;
    float* w_nv  = ws + WS_NV_OFF;
    float* w_nes = ws + WS_NES_OFF;
    float* w_nev = ws + WS_NEV_OFF;

    float* out  = (float*)d_out;
    float* o_ns = out + OUT_NS_OFF;
    float* o_nv = out + OUT_NV_OFF;
    float* o_es = out + OUT_ES_OFF;
    float* o_ev = out + OUT_EV_OFF;

    // 1) zero scatter accumulators (n_es + n_ev = N*64 floats)
    zero_kernel<<<12500, 256, 0, stream>>>(w_nes, (long)N_NODES * 64);
    // 2) node input projections (persistent waves, weights in VGPRs)
    node_in_s_kernel<<<196, 128, 0, stream>>>(node_s, ns_w, ns_b, w_ns);
    node_in_v_kernel<<<196, 128, 0, stream>>>(node_v, nv_w, w_nv);
    // 3) edge update + scatter (100000 16-edge tiles; 20000 waves x 5 tiles)
    edge_kernel<<<2500, 256, 0, stream>>>(
        edge_s, edge_v, dist, vctr, src, dst, w_ns, w_nv, en_w, en_b, tp_w, tp_b,
        gw1, gb1, gw2, gb2, tv_w, tv_b, w_nes, w_nev, o_es, o_ev);
    // 4) node updates + norms (persistent: 625 blocks x 5 tiles)
    node_out_kernel<<<625, 128, 0, stream>>>(node_s, node_v, w_nv, w_nes, w_nev,
                                             onv_w, ow1, ob1, ow2, ob2, gamma,
                                             beta, cscale, o_ns, o_nv);
}